// GINE_34780645163305
// MI455X (gfx1250) — compile-verified
//
#include <hip/hip_runtime.h>
#include <hip/hip_bf16.h>
#include <stdint.h>

// ---------------------------------------------------------------------------
// GIN (3 layers) + BN + global_add_pool + MLP head, fp32, MI455X (gfx1250).
// GEMMs use V_WMMA_F32_16X16X4_F32 (fp32 tensor path; exact vs fp32 ref).
// Edge aggregation = float4 gather + global fp32 atomic scatter (memory-bound
// critical path; feature matrices fit in 192MB L2).
// ---------------------------------------------------------------------------

typedef float v2f __attribute__((ext_vector_type(2)));
typedef float v8f __attribute__((ext_vector_type(8)));

// ---------------- elementwise helpers ----------------

__global__ void zero_kernel(float* __restrict__ p, long long n) {
    long long i = (long long)blockIdx.x * blockDim.x + threadIdx.x;
    if (i < n) p[i] = 0.f;
}

__global__ void copy4_kernel(float* __restrict__ dst, const float* __restrict__ src, long long n4) {
    long long i = (long long)blockIdx.x * blockDim.x + threadIdx.x;
    if (i < n4) ((float4*)dst)[i] = ((const float4*)src)[i];
}

// ---------------- edge scatter: agg[dst] += x[src] ----------------
// one thread per (edge, 4-feature chunk)
__global__ void scatter_add_kernel(float* __restrict__ agg, const float* __restrict__ x,
                                   const int* __restrict__ src, const int* __restrict__ dst,
                                   long long nE, int F) {
    const int chunks = F >> 2;
    long long i = (long long)blockIdx.x * blockDim.x + threadIdx.x;
    long long total = nE * (long long)chunks;
    if (i >= total) return;
    long long e = i / chunks;
    int c = (int)(i - e * chunks) << 2;
    int s = src[e];
    int d = dst[e];
    float4 v = *(const float4*)(x + (size_t)s * F + c);
    float* o = agg + (size_t)d * F + c;
    atomicAdd(o + 0, v.x);
    atomicAdd(o + 1, v.y);
    atomicAdd(o + 2, v.z);
    atomicAdd(o + 3, v.w);
}

// ---------------- WMMA fp32 GEMM: out = act(in[nrows x Fi] @ W[Fi x Fo] + b) ----------------
// one wave per 16x16 output tile, K stepped by 4 via v_wmma_f32_16x16x4_f32.
// VGPR layout per ISA: 32-bit A 16x4 -> lane<16 holds K={k0,k0+1}, lane>=16 holds K={k0+2,k0+3};
// B 4x16 mirrors it; C/D 16x16 -> VGPR r: lanes0-15 M=r, lanes16-31 M=r+8.
__global__ __launch_bounds__(256) void gemm_wmma_f32(
    float* __restrict__ out, const float* __restrict__ in,
    const float* __restrict__ W, const float* __restrict__ bias,
    int nrows, int Fi, int Fo, int relu)
{
    const int lane = threadIdx.x & 31;
    const int wave = threadIdx.x >> 5;
    const int ntilesN = Fo >> 4;
    const int mtiles  = nrows >> 4;                 // nrows % 16 == 0
    const long long total = (long long)mtiles * ntilesN;
    long long wid = (long long)blockIdx.x * 8 + wave;
    if (wid >= total) return;                       // wave-uniform: EXEC stays all-1s
    const int mt = (int)(wid / ntilesN);
    const int nt = (int)(wid % ntilesN);

    const int half = lane >> 4;                     // 0: lanes 0-15, 1: lanes 16-31
    const int l16  = lane & 15;
    const int arow = mt * 16 + l16;                 // A-fragment row (M)
    const int bcol = nt * 16 + l16;                 // B-fragment col (N)
    const int koff = half * 2;                      // K sub-offset per half-wave

    v8f acc = {};
    const float* arowp = in + (size_t)arow * Fi;
    for (int k0 = 0; k0 < Fi; k0 += 4) {
        v2f a = *(const v2f*)(arowp + k0 + koff);   // {K=k0+koff, K=k0+koff+1}
        v2f b;
        b.x = W[(size_t)(k0 + koff)     * Fo + bcol];
        b.y = W[(size_t)(k0 + koff + 1) * Fo + bcol];
        acc = __builtin_amdgcn_wmma_f32_16x16x4_f32(
            /*neg_a=*/false, a, /*neg_b=*/false, b,
            /*c_mod=*/(short)0, acc, /*reuse_a=*/false, /*reuse_b=*/false);
    }

    const float bv = bias[bcol];
#pragma unroll
    for (int r = 0; r < 8; ++r) {
        int row = mt * 16 + half * 8 + r;
        float v = acc[r] + bv;
        if (relu) v = fmaxf(v, 0.f);
        out[(size_t)row * Fo + bcol] = v;
    }
}

// ---------------- BN statistics: per-column sum & sum-of-squares ----------------
// block handles a slab of rows for all F columns (F in {32,64,128}, power of 2)
__global__ __launch_bounds__(256) void stats_kernel(
    const float* __restrict__ y, int nrows, int F, int rowsPerBlock,
    float* __restrict__ colsum, float* __restrict__ colsq)
{
    __shared__ float ssum[256];
    __shared__ float ssq[256];
    const int tid = threadIdx.x;
    const int col = tid & (F - 1);
    const int rstep = 256 / F;
    const int rbeg = blockIdx.x * rowsPerBlock;
    const int rend = min(nrows, rbeg + rowsPerBlock);
    float s = 0.f, s2 = 0.f;
    for (int r = rbeg + tid / F; r < rend; r += rstep) {
        float v = y[(size_t)r * F + col];
        s += v; s2 += v * v;
    }
    ssum[tid] = s; ssq[tid] = s2;
    __syncthreads();
    for (int stride = 128; stride >= F; stride >>= 1) {
        if (tid < stride) { ssum[tid] += ssum[tid + stride]; ssq[tid] += ssq[tid + stride]; }
        __syncthreads();
    }
    if (tid < F) {
        atomicAdd(&colsum[col], ssum[tid]);
        atomicAdd(&colsq[col],  ssq[tid]);
    }
}

// ---------------- BN apply (in place): y = (y-mu)*rsqrt(var+eps)*g + be ----------------
__global__ void bn_apply_kernel(float* __restrict__ y,
                                const float* __restrict__ colsum, const float* __restrict__ colsq,
                                const float* __restrict__ g, const float* __restrict__ be,
                                long long nrows, int F, float invN)
{
    long long i = (long long)blockIdx.x * blockDim.x + threadIdx.x;
    long long total = nrows * (long long)F;
    if (i >= total) return;
    int col = (int)(i % F);
    float mu  = colsum[col] * invN;
    float var = fmaxf(colsq[col] * invN - mu * mu, 0.f);
    float sc  = rsqrtf(var + 1e-5f) * g[col];
    y[i] = (y[i] - mu) * sc + be[col];
}

// ---------------- global_add_pool: pool[batch[n]] += z[n] ----------------
__global__ void pool_kernel(float* __restrict__ pool, const float* __restrict__ z,
                            const int* __restrict__ batch, long long nrows, int F) {
    long long i = (long long)blockIdx.x * blockDim.x + threadIdx.x;
    long long total = nrows * (long long)F;
    if (i >= total) return;
    long long row = i / F;
    int col = (int)(i - row * F);
    atomicAdd(&pool[(size_t)batch[row] * F + col], z[i]);
}

// ---------------- small dense head (G=1024 rows; negligible cost) ----------------
__global__ void fc_kernel(float* __restrict__ out, const float* __restrict__ in,
                          const float* __restrict__ W, const float* __restrict__ b,
                          int rows, int Fi, int Fo, int relu) {
    int i = blockIdx.x * blockDim.x + threadIdx.x;
    if (i >= rows * Fo) return;
    int r = i / Fo, c = i - r * Fo;
    float acc = b[c];
    const float* ir = in + (size_t)r * Fi;
    for (int k = 0; k < Fi; ++k) acc += ir[k] * W[(size_t)k * Fo + c];
    if (relu) acc = fmaxf(acc, 0.f);
    out[i] = acc;
}

// ---------------------------------------------------------------------------

static inline unsigned gridFor(long long n, int block) {
    return (unsigned)((n + block - 1) / block);
}

extern "C" void kernel_launch(void* const* d_in, const int* in_sizes, int n_in,
                              void* d_out, int out_size, void* d_ws, size_t ws_size,
                              hipStream_t stream) {
    (void)n_in; (void)out_size; (void)ws_size;

    const float* x     = (const float*)d_in[0];
    const int*   eidx  = (const int*)d_in[1];
    const int*   batch = (const int*)d_in[2];
    const float* W1a = (const float*)d_in[3],  *b1a = (const float*)d_in[4];
    const float* W1b = (const float*)d_in[5],  *b1b = (const float*)d_in[6];
    const float* g1  = (const float*)d_in[7],  *be1 = (const float*)d_in[8];
    const float* W2a = (const float*)d_in[9],  *b2a = (const float*)d_in[10];
    const float* W2b = (const float*)d_in[11], *b2b = (const float*)d_in[12];
    const float* g2  = (const float*)d_in[13], *be2 = (const float*)d_in[14];
    const float* W3a = (const float*)d_in[15], *b3a = (const float*)d_in[16];
    const float* W3b = (const float*)d_in[17], *b3b = (const float*)d_in[18];
    const float* g3  = (const float*)d_in[19], *be3 = (const float*)d_in[20];
    const float* Wf1 = (const float*)d_in[21], *bf1 = (const float*)d_in[22];
    const float* Wf2 = (const float*)d_in[23], *bf2 = (const float*)d_in[24];
    const float* Wf3 = (const float*)d_in[25], *bf3 = (const float*)d_in[26];

    const long long Nn = in_sizes[0] / 64;       // 100000
    const long long Ee = in_sizes[1] / 2;        // 3200000
    const int G = 1024, Cc_ = 10;
    const int Nint = (int)Nn;
    const float invN = 1.0f / (float)Nn;

    const int* esrc = eidx;
    const int* edst = eidx + Ee;

    // workspace layout (floats)
    float* A    = (float*)d_ws;                  // N*128
    float* B    = A + (size_t)Nn * 128;          // N*128
    float* Cb   = B + (size_t)Nn * 128;          // N*128
    float* csum = Cb + (size_t)Nn * 128;         // 128
    float* csq  = csum + 128;                    // 128
    float* pool = csq + 128;                     // G*32
    float* f1   = pool + (size_t)G * 32;         // G*128
    float* f2   = f1 + (size_t)G * 128;          // G*64

    const int TB = 256;
    const int rowsPerBlock = 256;
    const unsigned statBlocks = gridFor(Nn, rowsPerBlock);

    // ---------------- Layer 1 (F=64 -> 128 -> 128) ----------------
    copy4_kernel<<<gridFor(Nn * 64 / 4, TB), TB, 0, stream>>>(A, x, Nn * 64 / 4);
    scatter_add_kernel<<<gridFor(Ee * (64 / 4), TB), TB, 0, stream>>>(A, x, esrc, edst, Ee, 64);
    {
        long long tiles = (Nn / 16) * (128 / 16);
        gemm_wmma_f32<<<gridFor(tiles, 8), TB, 0, stream>>>(B, A, W1a, b1a, Nint, 64, 128, 1);
        gemm_wmma_f32<<<gridFor(tiles, 8), TB, 0, stream>>>(A, B, W1b, b1b, Nint, 128, 128, 1);
    }
    zero_kernel<<<1, 256, 0, stream>>>(csum, 256);   // csum+csq contiguous
    stats_kernel<<<statBlocks, TB, 0, stream>>>(A, Nint, 128, rowsPerBlock, csum, csq);
    bn_apply_kernel<<<gridFor(Nn * 128, TB), TB, 0, stream>>>(A, csum, csq, g1, be1, Nn, 128, invN);

    // ---------------- Layer 2 (F=128 -> 64 -> 64) ----------------
    copy4_kernel<<<gridFor(Nn * 128 / 4, TB), TB, 0, stream>>>(B, A, Nn * 128 / 4);
    scatter_add_kernel<<<gridFor(Ee * (128 / 4), TB), TB, 0, stream>>>(B, A, esrc, edst, Ee, 128);
    {
        long long tiles = (Nn / 16) * (64 / 16);
        gemm_wmma_f32<<<gridFor(tiles, 8), TB, 0, stream>>>(Cb, B, W2a, b2a, Nint, 128, 64, 1);
        gemm_wmma_f32<<<gridFor(tiles, 8), TB, 0, stream>>>(B, Cb, W2b, b2b, Nint, 64, 64, 1);
    }
    zero_kernel<<<1, 256, 0, stream>>>(csum, 256);
    stats_kernel<<<statBlocks, TB, 0, stream>>>(B, Nint, 64, rowsPerBlock, csum, csq);
    bn_apply_kernel<<<gridFor(Nn * 64, TB), TB, 0, stream>>>(B, csum, csq, g2, be2, Nn, 64, invN);

    // ---------------- Layer 3 (F=64 -> 32 -> 32) ----------------
    copy4_kernel<<<gridFor(Nn * 64 / 4, TB), TB, 0, stream>>>(Cb, B, Nn * 64 / 4);
    scatter_add_kernel<<<gridFor(Ee * (64 / 4), TB), TB, 0, stream>>>(Cb, B, esrc, edst, Ee, 64);
    {
        long long tiles = (Nn / 16) * (32 / 16);
        gemm_wmma_f32<<<gridFor(tiles, 8), TB, 0, stream>>>(A, Cb, W3a, b3a, Nint, 64, 32, 1);
        gemm_wmma_f32<<<gridFor(tiles, 8), TB, 0, stream>>>(Cb, A, W3b, b3b, Nint, 32, 32, 1);
    }
    zero_kernel<<<1, 256, 0, stream>>>(csum, 256);
    stats_kernel<<<statBlocks, TB, 0, stream>>>(Cb, Nint, 32, rowsPerBlock, csum, csq);
    bn_apply_kernel<<<gridFor(Nn * 32, TB), TB, 0, stream>>>(Cb, csum, csq, g3, be3, Nn, 32, invN);

    // ---------------- pool + head ----------------
    zero_kernel<<<gridFor((long long)G * 32, TB), TB, 0, stream>>>(pool, (long long)G * 32);
    pool_kernel<<<gridFor(Nn * 32, TB), TB, 0, stream>>>(pool, Cb, batch, Nn, 32);

    fc_kernel<<<gridFor((long long)G * 128, TB), TB, 0, stream>>>(f1, pool, Wf1, bf1, G, 32, 128, 1);
    fc_kernel<<<gridFor((long long)G * 64, TB), TB, 0, stream>>>(f2, f1, Wf2, bf2, G, 128, 64, 1);
    fc_kernel<<<gridFor((long long)G * Cc_, TB), TB, 0, stream>>>((float*)d_out, f2, Wf3, bf3, G, 64, Cc_, 0);
}